// WaveletSparsityLoss_34746285424743
// MI455X (gfx1250) — compile-verified
//
#include <hip/hip_runtime.h>
#include <hip/hip_bf16.h>

// WaveletSparsityLoss for MI455X (gfx1250, wave32).
// Single streaming pass: 1 thread per 8x8 pixel block computes all 3 Haar
// levels in registers; per-coefficient weights folded so the final answer is
// just a global sum. Wave reduction via V_WMMA_F32_16X16X4_F32 (f32-exact).

typedef __attribute__((ext_vector_type(2))) float v2f;
typedef __attribute__((ext_vector_type(8))) float v8f;

// Folded constants: thr_s = (50 / 2^(3-s)) / 255 ; scale_s = w_s / (3 * N_s)
#define THR1 0.049019607843137254f   /* 12.5/255  */
#define THR2 0.098039215686274508f   /* 25/255    */
#define THR3 0.196078431372549020f   /* 50/255    */
#define SCL1 (1.0f / 75497472.0f)    /* (1/3)/(3*32*512*512) */
#define SCL2 (1.0f / 12582912.0f)    /* (1/2)/(3*32*256*256) */
#define SCL3 (1.0f / 1572864.0f)     /* (1/1)/(3*32*128*128) */

// Full-wave (32-lane) sum via two chained v_wmma_f32_16x16x4_f32.
// A-matrix 16x4 f32 layout: lanes 0-15 hold K=0 (x) / K=1 (y) for M=lane,
// lanes 16-31 hold K=2 (x) / K=3 (y) for M=lane-16.  B = ones(4x16).
// WMMA1: D[m][n] = p_m + p_{m+16}  (rowsums, replicated over N)
//   lane l, vgpr v holds D[v + 8*(l/16)][l%16]  -> s = sum_v d[v] gives
//   S_lo (lanes 0-15) = sum rowsum[0..7], S_hi (lanes 16-31) = sum rowsum[8..15]
// WMMA2 on s: every element = S_lo + S_hi = full wave sum.
__device__ __forceinline__ float wave_reduce_f32_wmma(float p) {
  v2f a;    a.x = p;    a.y = 0.0f;
  v2f ones; ones.x = 1.0f; ones.y = 1.0f;
  v8f c = {};
  v8f d = __builtin_amdgcn_wmma_f32_16x16x4_f32(
      /*neg_a=*/false, a, /*neg_b=*/false, ones,
      /*c_mod=*/(short)0, c, /*reuse_a=*/false, /*reuse_b=*/false);
  float s = ((d[0] + d[1]) + (d[2] + d[3])) + ((d[4] + d[5]) + (d[6] + d[7]));
  v2f a2; a2.x = s; a2.y = 0.0f;
  v8f d2 = __builtin_amdgcn_wmma_f32_16x16x4_f32(
      false, a2, false, ones, (short)0, c, false, false);
  return d2[0];
}

// One 2x2 Haar quad: returns weighted clipped-band contribution, outputs cA.
__device__ __forceinline__ float haar_quad(float a, float b, float c, float d,
                                           float thr, float scl, float& cA) {
  cA        = (a + b + c + d) * 0.5f;
  float cH  = (a + b - c - d) * 0.5f;
  float cV  = (a - b + c - d) * 0.5f;
  float cD  = (a - b - c + d) * 0.5f;
  return scl * (fminf(fabsf(cH), thr) +
                fminf(fabsf(cV), thr) +
                fminf(fabsf(cD), thr));
}

// Grid: 2048 blocks x 256 threads = 524288 threads = 32 images * (128x128)
// 8x8 blocks.  Each thread streams its 8x8 tile (256 B) with float4 loads.
__global__ void __launch_bounds__(256)
haar_loss_main(const float* __restrict__ pred, float* __restrict__ partials) {
  const int T   = blockIdx.x * 256 + threadIdx.x;
  const int bx  = T & 127;          // 8-wide block column
  const int by  = (T >> 7) & 127;   // 8-tall block row
  const int img = T >> 14;          // image index

  const float* base = pred + ((size_t)img << 20) + ((size_t)by << 13) + (bx << 3);

  float acc = 0.0f;
  float cA1[4][4];

  // ---- Level 1: stream 4 row-pairs ----
  #pragma unroll
  for (int rp = 0; rp < 4; ++rp) {
    if (rp == 0) {
      // prefetch the back half of this 8-row tile (always in-bounds)
      __builtin_prefetch(base + 4 * 1024, 0, 0);
      __builtin_prefetch(base + 6 * 1024, 0, 0);
    }
    const float* r0 = base + (size_t)(2 * rp) * 1024;
    const float4 r0a = *(const float4*)(r0);
    const float4 r0b = *(const float4*)(r0 + 4);
    const float4 r1a = *(const float4*)(r0 + 1024);
    const float4 r1b = *(const float4*)(r0 + 1028);

    acc += haar_quad(r0a.x, r0a.y, r1a.x, r1a.y, THR1, SCL1, cA1[rp][0]);
    acc += haar_quad(r0a.z, r0a.w, r1a.z, r1a.w, THR1, SCL1, cA1[rp][1]);
    acc += haar_quad(r0b.x, r0b.y, r1b.x, r1b.y, THR1, SCL1, cA1[rp][2]);
    acc += haar_quad(r0b.z, r0b.w, r1b.z, r1b.w, THR1, SCL1, cA1[rp][3]);
  }

  // ---- Level 2: 4x4 cA1 -> 2x2 cA2 ----
  float cA2[2][2];
  #pragma unroll
  for (int r = 0; r < 2; ++r)
    #pragma unroll
    for (int c = 0; c < 2; ++c)
      acc += haar_quad(cA1[2 * r][2 * c],     cA1[2 * r][2 * c + 1],
                       cA1[2 * r + 1][2 * c], cA1[2 * r + 1][2 * c + 1],
                       THR2, SCL2, cA2[r][c]);

  // ---- Level 3: 2x2 cA2 -> 1 quad ----
  float cA3;
  acc += haar_quad(cA2[0][0], cA2[0][1], cA2[1][0], cA2[1][1],
                   THR3, SCL3, cA3);
  (void)cA3;

  // ---- Wave reduce (WMMA, f32-exact), then block reduce via LDS ----
  float wsum = wave_reduce_f32_wmma(acc);

  __shared__ float lds[8];
  const int lane = threadIdx.x & 31;
  const int wv   = threadIdx.x >> 5;
  if (lane == 0) lds[wv] = wsum;
  __syncthreads();
  if (threadIdx.x == 0) {
    float t = 0.0f;
    #pragma unroll
    for (int i = 0; i < 8; ++i) t += lds[i];
    partials[blockIdx.x] = t;
  }
}

// Single wave folds the 2048 block partials (deterministic order, no float
// atomics) and finishes with the same WMMA wave reduction.
__global__ void __launch_bounds__(32)
haar_loss_final(const float* __restrict__ partials, float* __restrict__ out,
                int n) {
  float p = 0.0f;
  for (int i = (int)threadIdx.x; i < n; i += 32) p += partials[i];
  float tot = wave_reduce_f32_wmma(p);
  if (threadIdx.x == 0) out[0] = tot;
}

extern "C" void kernel_launch(void* const* d_in, const int* in_sizes, int n_in,
                              void* d_out, int out_size, void* d_ws, size_t ws_size,
                              hipStream_t stream) {
  const float* pred = (const float*)d_in[0];
  float* out        = (float*)d_out;
  float* partials   = (float*)d_ws;   // 2048 floats = 8 KB scratch

  const int nBlocks = 2048;           // 524288 threads, 64 px each = 32*1024*1024
  haar_loss_main<<<nBlocks, 256, 0, stream>>>(pred, partials);
  haar_loss_final<<<1, 32, 0, stream>>>(partials, out, nBlocks);
}